// DeformConv1d_28707561406943
// MI455X (gfx1250) — compile-verified
//
#include <hip/hip_runtime.h>

typedef __attribute__((ext_vector_type(2))) float v2f;
typedef __attribute__((ext_vector_type(8))) float v8f;

#define B_   16
#define C_   64
#define L_   8192
#define O_   64
#define K_   3
#define R_   (C_ * K_)   // 192 = contraction length
#define TN   64          // output positions per block
#define SSTR 196         // padded r-stride (192 + 4) to break LDS bank conflicts

__global__ __launch_bounds__(256)
void deform_conv1d_wmma(const float* __restrict__ x,
                        const float* __restrict__ offsets,
                        const float* __restrict__ weight,
                        const float* __restrict__ bias,
                        float* __restrict__ out) {
  __shared__ float lds_w[O_ * SSTR];   // W[f][r], r-contiguous
  __shared__ float lds_s[TN * SSTR];   // S[o_local][r], r-contiguous
  __shared__ float lds_p0[TN * K_];    // w0 per (o,k)
  __shared__ float lds_p1[TN * K_];    // w1 per (o,k)
  __shared__ int   lds_u [TN * K_];    // U  per (o,k)

  const int tid = threadIdx.x;
  const int ob  = blockIdx.x * TN;     // first output position of this block
  const int b   = blockIdx.y;

  // ---- Phase 0: weight -> LDS (global layout [f][c][k] is already r-contiguous)
  for (int i = tid; i < (O_ * R_) / 4; i += 256) {
    const int f  = i / (R_ / 4);
    const int rq = i - f * (R_ / 4);
    const float4 w4 = ((const float4*)weight)[f * (R_ / 4) + rq];
    float* d = &lds_w[f * SSTR + rq * 4];
    d[0] = w4.x; d[1] = w4.y; d[2] = w4.z; d[3] = w4.w;
  }

  // ---- Phase 1a: interpolation params, one thread per (o_local, k)
  if (tid < TN * K_) {
    const int o_l = tid / K_;
    const int k   = tid - o_l * K_;
    const int o   = ob + o_l;
    const float off = offsets[(b * L_ + o) * K_ + k];
    float T = (float)(o + k) + off;
    T = fminf(fmaxf(T, (float)o), (float)(o + K_ - 1));   // clamp to receptive field
    int U = (int)floorf(T);
    U = min(max(U, 0), L_);                               // Lp-2 == L_
    const float Uf = (float)U;
    lds_u [tid] = U;
    lds_p0[tid] = fmaxf(0.f, 1.f - fabsf(Uf - T));
    lds_p1[tid] = fmaxf(0.f, 1.f - fabsf(Uf + 1.f - T));
  }
  __syncthreads();

  // ---- Phase 1b: gather + lerp -> S[o][r] in LDS (reflect 'same' padding, pad=1 each side)
  {
    const int o_l = tid & (TN - 1);
    const int c0  = tid >> 6;                              // 0..3
    const float* xb = x + (size_t)b * C_ * L_;
    int   j0[K_], j1[K_];
    float w0v[K_], w1v[K_];
#pragma unroll
    for (int k = 0; k < K_; ++k) {
      const int U = lds_u[o_l * K_ + k];
      w0v[k] = lds_p0[o_l * K_ + k];
      w1v[k] = lds_p1[o_l * K_ + k];
      int a = U - 1;                                       // x_pad[U] -> x[U-1], reflected
      a = (a < 0) ? -a : a;
      a = (a > L_ - 1) ? (2 * (L_ - 1) - a) : a;
      int c = U;                                           // x_pad[U+1] -> x[U], reflected
      c = (c > L_ - 1) ? (2 * (L_ - 1) - c) : c;
      j0[k] = a; j1[k] = c;
    }
    for (int c = c0; c < C_; c += 4) {
      const float* xc = xb + (size_t)c * L_;
#pragma unroll
      for (int k = 0; k < K_; ++k) {
        lds_s[o_l * SSTR + c * K_ + k] = w0v[k] * xc[j0[k]] + w1v[k] * xc[j1[k]];
      }
    }
  }
  __syncthreads();

  // ---- Phase 2: 64x64 tile GEMM via V_WMMA_F32_16X16X4_F32
  const int wave = tid >> 5;            // 0..7
  const int lane = tid & 31;
  const int ln   = lane & 15;
  const int kh   = lane >> 4;           // K-half select (ISA A/B layout)

  const int m_tile = wave & 3;          // 4 tiles of f
  const int n0     = (wave >> 2) * 2;   // this wave covers n-tiles n0, n0+1

  const float* aw = &lds_w[(m_tile * 16 + ln) * SSTR + kh * 2];
  const float* bp0 = &lds_s[((n0    ) * 16 + ln) * SSTR + kh * 2];
  const float* bp1 = &lds_s[((n0 + 1) * 16 + ln) * SSTR + kh * 2];

  v8f acc0 = {};
  v8f acc1 = {};
#pragma unroll 4
  for (int r4 = 0; r4 < R_; r4 += 4) {
    const v2f a  = *(const v2f*)(aw  + r4);
    const v2f b0 = *(const v2f*)(bp0 + r4);
    const v2f b1 = *(const v2f*)(bp1 + r4);
    acc0 = __builtin_amdgcn_wmma_f32_16x16x4_f32(false, a, false, b0, (short)0, acc0, false, false);
    acc1 = __builtin_amdgcn_wmma_f32_16x16x4_f32(false, a, false, b1, (short)0, acc1, false, false);
  }

  // ---- Epilogue: D layout -> out[b][f][o] + bias
  const int og0 = ob + n0 * 16 + ln;
  const int og1 = og0 + 16;
#pragma unroll
  for (int v = 0; v < 8; ++v) {
    const int f = m_tile * 16 + v + kh * 8;
    const float bv = bias[f];
    const size_t base = ((size_t)b * O_ + f) * L_;
    out[base + og0] = acc0[v] + bv;
    out[base + og1] = acc1[v] + bv;
  }
}

extern "C" void kernel_launch(void* const* d_in, const int* in_sizes, int n_in,
                              void* d_out, int out_size, void* d_ws, size_t ws_size,
                              hipStream_t stream) {
  const float* x       = (const float*)d_in[0];
  const float* offsets = (const float*)d_in[1];
  const float* weight  = (const float*)d_in[2];
  const float* bias    = (const float*)d_in[3];
  float* out = (float*)d_out;

  dim3 grid(L_ / TN, B_);  // 128 x 16 blocks, 256 threads each
  deform_conv1d_wmma<<<grid, 256, 0, stream>>>(x, offsets, weight, bias, out);
}